// ProposalLayer_33964601377079
// MI455X (gfx1250) — compile-verified
//
#include <hip/hip_runtime.h>
#include <math.h>

// ---------------- problem constants ----------------
constexpr int kB    = 32;
constexpr int kA    = 9;
constexpr int kH    = 100;
constexpr int kW    = 152;
constexpr int kHW   = kH * kW;          // 15200
constexpr int kN    = kA * kHW;         // 136800 anchors per batch
constexpr int kPRE  = 2000;
constexpr int kPOST = 300;
constexpr float kNMS = 0.7f;
constexpr int kNT   = kPRE / 16;        // 125 16x16 tiles per side
constexpr int kMW   = 128;              // u16 mask words per row (125 used, padded)

// precomputed generate_anchors(16, [0.5,1,2], [8,16,32])
__device__ __constant__ float kAnch[kA][4] = {
    {-84.f,  -40.f,  99.f,  55.f},
    {-176.f, -88.f,  191.f, 103.f},
    {-360.f, -184.f, 375.f, 199.f},
    {-56.f,  -56.f,  71.f,  71.f},
    {-120.f, -120.f, 135.f, 135.f},
    {-248.f, -248.f, 263.f, 263.f},
    {-36.f,  -80.f,  51.f,  95.f},
    {-80.f,  -168.f, 95.f,  183.f},
    {-168.f, -344.f, 183.f, 359.f}};

typedef float v2f __attribute__((ext_vector_type(2)));
typedef float v8f __attribute__((ext_vector_type(8)));
typedef unsigned int       u32;
typedef unsigned long long u64;

// workspace layout (bytes). mask region aliases sb region (sb dead after select).
constexpr size_t SB_BYTES  = (size_t)kB * kN * sizeof(u32);         // 17,510,400
constexpr size_t SEL_BYTES = (size_t)kB * 2048 * sizeof(u64);       //    524,288
constexpr size_t SEL_OFF   = SB_BYTES;
constexpr size_t BOX_OFF   = SEL_OFF + SEL_BYTES;

static __device__ __forceinline__ u32 wave_ballot32(bool c) {
#if __has_builtin(__builtin_amdgcn_ballot_w32)
    return __builtin_amdgcn_ballot_w32(c);
#else
    return (u32)__ballot(c);
#endif
}

// ---------------- K1: objectness scores -> orderable u32 keys ----------------
__global__ __launch_bounds__(256) void score_bits_kernel(const float* __restrict__ scores,
                                                         u32* __restrict__ sb) {
    size_t tid = (size_t)blockIdx.x * blockDim.x + threadIdx.x;
    if (tid >= (size_t)kB * kN) return;
    size_t b = tid / kN;
    size_t i = tid - b * kN;                    // i = a*HW + s  (channel-major)
    // scores[b][kA + a][s] : base + b*2A*HW + A*HW + i
    float v = scores[b * (size_t)(2 * kA) * kHW + (size_t)kA * kHW + i];
    sb[tid] = __float_as_uint(v);               // scores >= 0 -> bits order == float order
}

// ---------------- K2: per-batch radix select of exact top-2000 ----------------
__global__ __launch_bounds__(256) void radix_select_kernel(const u32* __restrict__ sb,
                                                           u64* __restrict__ sel) {
    const int b = blockIdx.x;
    const u32* S = sb + (size_t)b * kN;
    __shared__ u32 hist[256];
    __shared__ u32 sDigit, sNeed;
    u32 prefix = 0, prefmask = 0, need = kPRE;

    for (int pass = 0; pass < 4; ++pass) {
        const int shift = 24 - 8 * pass;
        for (int i = threadIdx.x; i < 256; i += blockDim.x) hist[i] = 0;
        __syncthreads();
        for (int i = threadIdx.x; i < kN; i += blockDim.x) {
            u32 v = S[i];
            if ((v & prefmask) == prefix) atomicAdd(&hist[(v >> shift) & 0xFFu], 1u);
        }
        __syncthreads();
        if (threadIdx.x == 0) {
            u32 cum = 0; int d = 255;
            for (; d > 0; --d) {
                u32 h = hist[d];
                if (cum + h >= need) break;
                cum += h;
            }
            sDigit = (u32)d;
            sNeed  = need - cum;
        }
        __syncthreads();
        prefix   |= sDigit << shift;
        prefmask |= 0xFFu << shift;
        need      = sNeed;
        __syncthreads();
    }

    const u32 T = prefix;                 // exact value of the 2000th-largest score
    const u32 G = kPRE - need;            // count strictly greater than T
    __shared__ u32 cntG, cntE;
    if (threadIdx.x == 0) { cntG = 0; cntE = 0; }
    __syncthreads();

    u64* out = sel + (size_t)b * 2048;
    for (int i = threadIdx.x; i < kN; i += blockDim.x) {
        u32 v = S[i];
        if (v < T) continue;
        u64 key = ((u64)v << 32) | (u64)(0xFFFFFFFFu - (u32)i);  // desc score, asc index
        if (v > T) {
            u32 s0 = atomicAdd(&cntG, 1u);
            out[s0] = key;
        } else {
            u32 e = atomicAdd(&cntE, 1u);
            if (e < need) out[G + e] = key;
        }
    }
    for (int i = kPRE + threadIdx.x; i < 2048; i += blockDim.x) out[i] = 0ull;
}

// ---------------- K3: bitonic sort 2048 keys + decode/clip selected boxes ----------------
__global__ __launch_bounds__(1024) void sort_decode_kernel(const u64* __restrict__ sel,
                                                           const float* __restrict__ deltas,
                                                           const float* __restrict__ img_info,
                                                           float4* __restrict__ boxes) {
    __shared__ u64 key[2048];
    const int b = blockIdx.x, t = threadIdx.x;
    key[t]        = sel[(size_t)b * 2048 + t];
    key[t + 1024] = sel[(size_t)b * 2048 + t + 1024];

    for (int k = 2; k <= 2048; k <<= 1) {
        for (int j = k >> 1; j > 0; j >>= 1) {
            __syncthreads();
            int i = ((t & ~(j - 1)) << 1) | (t & (j - 1));
            int p = i | j;
            bool up = (i & k) == 0;                  // descending overall
            u64 a = key[i], c = key[p];
            bool sw = up ? (a < c) : (a > c);
            if (sw) { key[i] = c; key[p] = a; }
        }
    }
    __syncthreads();

    const float hmax = img_info[b * 3 + 0] - 1.0f;
    const float wmax = img_info[b * 3 + 1] - 1.0f;
    for (int r = t; r < kPRE; r += 1024) {
        u64 kk = key[r];
        u32 p = 0xFFFFFFFFu - (u32)(kk & 0xFFFFFFFFull);   // p = a*HW + s
        int a  = (int)(p / kHW);
        int s  = (int)(p - (u32)a * kHW);
        int hh = s / kW;
        int ww = s - hh * kW;

        float x1 = kAnch[a][0] + ww * 16.f;
        float y1 = kAnch[a][1] + hh * 16.f;
        float x2 = kAnch[a][2] + ww * 16.f;
        float y2 = kAnch[a][3] + hh * 16.f;
        float wA = x2 - x1 + 1.f, hA = y2 - y1 + 1.f;
        float cx = x1 + 0.5f * wA, cy = y1 + 0.5f * hA;

        const float* dp = deltas + ((size_t)b * (4 * kA) + (size_t)(4 * a)) * kHW + s;
        float dx = dp[0], dy = dp[kHW], dw = dp[2 * kHW], dh = dp[3 * kHW];

        float px = dx * wA + cx, py = dy * hA + cy;
        float pw = expf(dw) * wA, ph = expf(dh) * hA;

        float bx1 = fminf(fmaxf(px - 0.5f * pw, 0.f), wmax);
        float by1 = fminf(fmaxf(py - 0.5f * ph, 0.f), hmax);
        float bx2 = fminf(fmaxf(px + 0.5f * pw, 0.f), wmax);
        float by2 = fminf(fmaxf(py + 0.5f * ph, 0.f), hmax);
        boxes[(size_t)b * kPRE + r] = make_float4(bx1, by1, bx2, by2);
    }
}

// ---------------- K4: IoU suppression bit-masks, one 16x16 tile per wave32 ----------------
// area_i + area_j outer sum for the whole tile via V_WMMA_F32_16X16X4_F32.
__global__ __launch_bounds__(128) void nms_mask_kernel(const float4* __restrict__ boxes,
                                                       unsigned short* __restrict__ mask) {
    const int wave = threadIdx.x >> 5;
    const int lane = threadIdx.x & 31;
    const int c = blockIdx.x * 4 + wave;          // column tile
    const int r = blockIdx.y;                     // row tile
    const int b = blockIdx.z;
    if (c >= kNT || c < r) return;                // wave-uniform: EXEC stays all-ones

    const float4* BB = boxes + (size_t)b * kPRE;
    const int rbase = r * 16, cbase = c * 16;

    // column box for this lane's N = lane%16; also used for B-operand area (lanes<16)
    float4 cb = BB[cbase + (lane & 15)];
    // 8 row boxes: M = v + 8*(lane/16)
    float4 rb[8];
    const int rr = rbase + ((lane >> 4) << 3);
#pragma unroll
    for (int v = 0; v < 8; ++v) rb[v] = BB[rr + v];
    // A-operand area: row M = lane (lanes 0..15)
    float4 ab = BB[rbase + (lane & 15)];

    const float areaRow = (ab.z - ab.x + 1.f) * (ab.w - ab.y + 1.f);
    const float areaCol = (cb.z - cb.x + 1.f) * (cb.w - cb.y + 1.f);

    v8f D;
#if __has_builtin(__builtin_amdgcn_wmma_f32_16x16x4_f32)
    // A (16x4): lanes 0-15 hold K=0,1 ; lanes 16-31 hold K=2,3 (zeros)
    // B (4x16): lanes 0-15 hold K=0,1 ; lanes 16-31 hold K=2,3 (zeros)
    // D[m][n] = areaRow[m]*1 + 1*areaCol[n]
    const bool lo = lane < 16;
    v2f Aop, Bop;
    Aop[0] = lo ? areaRow : 0.f;  Aop[1] = lo ? 1.f : 0.f;
    Bop[0] = lo ? 1.f : 0.f;      Bop[1] = lo ? areaCol : 0.f;
    v8f Cz = {};
    D = __builtin_amdgcn_wmma_f32_16x16x4_f32(false, Aop, false, Bop,
                                              (short)0, Cz, false, false);
#else
    // fallback: per-lane sum (same D layout)
#pragma unroll
    for (int v = 0; v < 8; ++v)
        D[v] = (rb[v].z - rb[v].x + 1.f) * (rb[v].w - rb[v].y + 1.f) + areaCol;
#endif

#pragma unroll
    for (int v = 0; v < 8; ++v) {
        const int gr = rbase + v + ((lane >> 4) << 3);
        const int gc = cbase + (lane & 15);
        float iw = fminf(rb[v].z, cb.z) - fmaxf(rb[v].x, cb.x) + 1.f;
        float ih = fminf(rb[v].w, cb.w) - fmaxf(rb[v].y, cb.y) + 1.f;
        iw = fmaxf(iw, 0.f);
        ih = fmaxf(ih, 0.f);
        const float inter = iw * ih;
        // inter/(sum-inter) > t  <=>  (1+t)*inter > t*sum   (union > 0)
        const bool cond = (gc > gr) && ((1.f + kNMS) * inter > kNMS * D[v]);
        const u32 bal = wave_ballot32(cond);      // bits 0-15: row v ; 16-31: row v+8
        if (lane == 0) {
            mask[((size_t)b * kPRE + rbase + v) * kMW + c]     = (unsigned short)(bal & 0xFFFFu);
            mask[((size_t)b * kPRE + rbase + v + 8) * kMW + c] = (unsigned short)(bal >> 16);
        }
    }
}

// ---------------- K5: sequential greedy scan + output ----------------
__global__ __launch_bounds__(128) void nms_scan_kernel(const u32* __restrict__ mask32,
                                                       const float4* __restrict__ boxes,
                                                       float* __restrict__ out) {
    __shared__ int keepIdx[kPOST];
    __shared__ int keepCnt;
    const int b = blockIdx.x;

    if (threadIdx.x < 32) {
        const int lane = threadIdx.x;
        u32 rem0 = 0, rem1 = 0;                    // suppressed bits: words lane, lane+32
        int cnt = 0;
        const u32* base = mask32 + (size_t)b * kPRE * (kMW / 2);
        for (int i = 0; i < kPRE; ++i) {
            if (i + 1 < kPRE)                      // gfx1250 global_prefetch of next row
                __builtin_prefetch((const void*)(base + (size_t)(i + 1) * (kMW / 2) + lane), 0, 0);
            const int wi = i >> 5;
            u32 myw = (wi < 32) ? rem0 : rem1;
            u32 ow  = __shfl(myw, wi & 31, 32);
            if (!((ow >> (i & 31)) & 1u)) {        // box i survives
                if (lane == 0) keepIdx[cnt] = i;
                ++cnt;
                if (cnt >= kPOST) break;
                const u32* row = base + (size_t)i * (kMW / 2);
                const int ci = i >> 4;             // first valid u16 chunk for this row
#pragma unroll
                for (int q = 0; q < 2; ++q) {
                    const int w  = lane + q * 32;
                    const int c0 = 2 * w, c1 = 2 * w + 1;
                    u32 m = 0;
                    if (c0 >= ci && c0 <= kNT - 1) m |= 0x0000FFFFu;
                    if (c1 >= ci && c1 <= kNT - 1) m |= 0xFFFF0000u;
                    if (m) {
                        u32 val = row[w] & m;
                        if (q == 0) rem0 |= val; else rem1 |= val;
                    }
                }
            }
        }
        if (lane == 0) keepCnt = cnt;
    }
    __syncthreads();

    const int cnt = keepCnt;
    for (int t = threadIdx.x; t < kPOST; t += blockDim.x) {
        float* o = out + ((size_t)b * kPOST + t) * 5;
        if (t < cnt) {
            float4 bx = boxes[(size_t)b * kPRE + keepIdx[t]];
            o[0] = (float)b; o[1] = bx.x; o[2] = bx.y; o[3] = bx.z; o[4] = bx.w;
        } else {
            o[0] = 0.f; o[1] = 0.f; o[2] = 0.f; o[3] = 0.f; o[4] = 0.f;
        }
    }
}

// ---------------- host-side launcher ----------------
extern "C" void kernel_launch(void* const* d_in, const int* in_sizes, int n_in,
                              void* d_out, int out_size, void* d_ws, size_t ws_size,
                              hipStream_t stream) {
    const float* scores   = (const float*)d_in[0];
    const float* deltas   = (const float*)d_in[1];
    const float* img_info = (const float*)d_in[2];
    // d_in[3] (gt_boxes) unused by the reference computation
    float* out = (float*)d_out;

    char* ws = (char*)d_ws;
    u32* sb                  = (u32*)ws;                     // B*N keys (reused as mask)
    u64* sel                 = (u64*)(ws + SEL_OFF);         // B*2048 sorted-key buffer
    float4* boxes            = (float4*)(ws + BOX_OFF);      // B*2000 decoded boxes
    unsigned short* mask16   = (unsigned short*)ws;          // aliases sb (dead after K2)
    const u32* mask32        = (const u32*)ws;

    {
        const size_t total = (size_t)kB * kN;
        dim3 g((unsigned)((total + 255) / 256)), blk(256);
        score_bits_kernel<<<g, blk, 0, stream>>>(scores, sb);
    }
    radix_select_kernel<<<dim3(kB), dim3(256), 0, stream>>>(sb, sel);
    sort_decode_kernel<<<dim3(kB), dim3(1024), 0, stream>>>(sel, deltas, img_info, boxes);
    nms_mask_kernel<<<dim3((kNT + 3) / 4, kNT, kB), dim3(128), 0, stream>>>(boxes, mask16);
    nms_scan_kernel<<<dim3(kB), dim3(128), 0, stream>>>(mask32, boxes, out);
}